// TaskAdaptiveGATLayer_3882650436801
// MI455X (gfx1250) — compile-verified
//
#include <hip/hip_runtime.h>
#include <math.h>

typedef float v2f __attribute__((ext_vector_type(2)));
typedef float v8f __attribute__((ext_vector_type(8)));

#define IN_DIM 128
#define HID 64

// ---------------------------------------------------------------------------
// K1: h = x @ W + b   via V_WMMA_F32_16X16X4_F32
// block = 128 threads (4 waves). Wave w computes the 16x16 tile at
// rows [blockIdx.x*16, +16), cols [w*16, +16).  K = 128 in steps of 4.
// A 16x4 layout (ISA 7.12.2): lanes 0-15: M=lane, VGPR0=K0,VGPR1=K1;
//                             lanes 16-31: M=lane-16, VGPR0=K2,VGPR1=K3.
// B 4x16: lanes 0-15: N=lane, VGPR0=K0,VGPR1=K1; lanes 16-31: N=lane-16, K2/K3.
// D: VGPR r -> row r + 8*(lane>=16), col lane%16.
// ---------------------------------------------------------------------------
__global__ void __launch_bounds__(128)
gat_gemm_wmma(const float* __restrict__ x, const float* __restrict__ W,
              const float* __restrict__ bias, float* __restrict__ h, int n_nodes) {
    const int lane  = threadIdx.x & 31;
    const int wave  = threadIdx.x >> 5;        // 0..3 -> N tile
    const int l     = lane & 15;
    const int lhalf = lane >> 4;               // 0 or 1
    const int m0    = blockIdx.x * 16;
    const int n0    = wave * 16;

    int arow = m0 + l;
    if (arow >= n_nodes) arow = n_nodes - 1;   // uniform-safe clamp (no divergence)
    const float* xr = x + (size_t)arow * IN_DIM;
    const float* wc = W + n0 + l;              // column n0+l of W [128 x 64]
    const int kb = 2 * lhalf;                  // per-lane K sub-offset

    v8f acc = {};
#pragma unroll
    for (int kk = 0; kk < IN_DIM; kk += 4) {
        v2f av, bv;
        av.x = xr[kk + kb];
        av.y = xr[kk + kb + 1];
        bv.x = wc[(size_t)(kk + kb) * HID];
        bv.y = wc[(size_t)(kk + kb + 1) * HID];
        acc = __builtin_amdgcn_wmma_f32_16x16x4_f32(
            false, av, false, bv, (short)0, acc, false, false);
    }

    const float bval = bias[n0 + l];
    const int mbase = m0 + 8 * lhalf;
#pragma unroll
    for (int r = 0; r < 8; ++r) {
        int m = mbase + r;
        if (m < n_nodes) h[(size_t)m * HID + n0 + l] = acc[r] + bval;
    }
}

// ---------------------------------------------------------------------------
// K2: per-node attention scalars s1 = h.a[:64], s2 = h.a[64:]; also
// (re)initialize segmax/-inf, segsum/0 and zero the output (must happen every
// call: atomics below accumulate and the harness replays the graph).
// One wave per node.
// ---------------------------------------------------------------------------
__global__ void __launch_bounds__(256)
gat_node_scores(const float* __restrict__ h, const float* __restrict__ a,
                float* __restrict__ s1, float* __restrict__ s2,
                float* __restrict__ segmax, float* __restrict__ segsum,
                float* __restrict__ out, int n_nodes) {
    const int node = blockIdx.x * 8 + (threadIdx.x >> 5);
    const int lane = threadIdx.x & 31;
    if (node >= n_nodes) return;

    const float* hp = h + (size_t)node * HID;
    const float v0 = hp[lane], v1 = hp[lane + 32];
    float p1 = v0 * a[lane]      + v1 * a[lane + 32];
    float p2 = v0 * a[64 + lane] + v1 * a[96 + lane];
#pragma unroll
    for (int off = 16; off > 0; off >>= 1) {
        p1 += __shfl_down(p1, off, 32);
        p2 += __shfl_down(p2, off, 32);
    }
    if (lane == 0) {
        s1[node] = p1;
        s2[node] = p2;
        segmax[node] = -INFINITY;
        segsum[node] = 0.0f;
    }
    out[(size_t)node * HID + lane]      = 0.0f;
    out[(size_t)node * HID + lane + 32] = 0.0f;
}

// float atomic max via sign-split integer atomics (init must be -inf)
__device__ __forceinline__ void atomicMaxF32(float* addr, float val) {
    if (val >= 0.0f) atomicMax((int*)addr, __float_as_int(val));
    else             atomicMin((unsigned int*)addr, __float_as_uint(val));
}

// ---------------------------------------------------------------------------
// K3: e = leaky_relu(s1[row] + s2[col]); store; segmax[row] = max(...)
// ---------------------------------------------------------------------------
__global__ void __launch_bounds__(256)
gat_edge_logit(const int* __restrict__ row, const int* __restrict__ col,
               const float* __restrict__ s1, const float* __restrict__ s2,
               float* __restrict__ evals, float* __restrict__ segmax, int n_edges) {
    const int i = blockIdx.x * 256 + threadIdx.x;
    if (i >= n_edges) return;
    const int r = row[i], c = col[i];
    float e = s1[r] + s2[c];
    e = (e > 0.0f) ? e : 0.2f * e;
    evals[i] = e;
    atomicMaxF32(segmax + r, e);
}

// ---------------------------------------------------------------------------
// K4: evals[i] = exp(evals[i] - segmax[row]); segsum[row] += evals[i]
// ---------------------------------------------------------------------------
__global__ void __launch_bounds__(256)
gat_edge_exp(const int* __restrict__ row, float* __restrict__ evals,
             const float* __restrict__ segmax, float* __restrict__ segsum, int n_edges) {
    const int i = blockIdx.x * 256 + threadIdx.x;
    if (i >= n_edges) return;
    const int r = row[i];
    const float ex = __expf(evals[i] - segmax[r]);
    evals[i] = ex;
    atomicAdd(segsum + r, ex);
}

// ---------------------------------------------------------------------------
// K5: out[row] += (evals[i] / (segsum[row]+1e-16)) * h[col]
// One wave per edge; each lane owns a float2 (64 floats per row).
// ---------------------------------------------------------------------------
__global__ void __launch_bounds__(256)
gat_scatter(const int* __restrict__ row, const int* __restrict__ col,
            const float* __restrict__ h, const float* __restrict__ evals,
            const float* __restrict__ segsum, float* __restrict__ out, int n_edges) {
    const int e = blockIdx.x * 8 + (threadIdx.x >> 5);
    const int lane = threadIdx.x & 31;
    if (e >= n_edges) return;
    const int r = row[e], c = col[e];
    const float alpha = evals[e] / (segsum[r] + 1e-16f);
    const float2 v = ((const float2*)(h + (size_t)c * HID))[lane];
    float* op = out + (size_t)r * HID + 2 * lane;
    atomicAdd(op,     alpha * v.x);
    atomicAdd(op + 1, alpha * v.y);
}

// ---------------------------------------------------------------------------
extern "C" void kernel_launch(void* const* d_in, const int* in_sizes, int n_in,
                              void* d_out, int out_size, void* d_ws, size_t ws_size,
                              hipStream_t stream) {
    const float* x    = (const float*)d_in[0];
    const int*   ei   = (const int*)d_in[1];   // [2, E]
    const float* a    = (const float*)d_in[2]; // 128 floats
    const float* W    = (const float*)d_in[3]; // [128, 64]
    const float* bias = (const float*)d_in[4]; // [64]
    float* out = (float*)d_out;

    const int N = in_sizes[0] / IN_DIM;
    const int E = in_sizes[1] / 2;
    const int* row = ei;
    const int* col = ei + E;

    // workspace layout (floats): h[N*64] | s1[N] | s2[N] | segmax[N] | segsum[N] | evals[E]
    float* ws     = (float*)d_ws;
    float* h      = ws;
    float* s1     = h + (size_t)N * HID;
    float* s2     = s1 + N;
    float* segmax = s2 + N;
    float* segsum = segmax + N;
    float* evals  = segsum + N;

    const int mtiles = (N + 15) / 16;
    gat_gemm_wmma<<<mtiles, 128, 0, stream>>>(x, W, bias, h, N);

    gat_node_scores<<<(N + 7) / 8, 256, 0, stream>>>(h, a, s1, s2, segmax, segsum, out, N);

    gat_edge_logit<<<(E + 255) / 256, 256, 0, stream>>>(row, col, s1, s2, evals, segmax, E);

    gat_edge_exp<<<(E + 255) / 256, 256, 0, stream>>>(row, evals, segmax, segsum, E);

    gat_scatter<<<(E + 7) / 8, 256, 0, stream>>>(row, col, h, evals, segsum, out, E);
}